// PhasorTransformer_84018150245086
// MI455X (gfx1250) — compile-verified
//
#include <hip/hip_runtime.h>
#include <hip/hip_bf16.h>

#define SEQ 2048
#define TPB 256
#define ROWS_PER_WG 4
#define PI_F 3.14159265358979323846f

typedef float v2f __attribute__((ext_vector_type(2)));
typedef float v8f __attribute__((ext_vector_type(8)));

// D = A(16x4) * B(4x16) + C, fp32 WMMA. All NEG/OPSEL modifiers zero
// (float WMMA only supports C-neg per ISA; negation is pre-baked into A).
__device__ __forceinline__ v8f wmma4(v2f a, v2f b, v8f c) {
  return __builtin_amdgcn_wmma_f32_16x16x4_f32(false, a, false, b, (short)0, c,
                                               false, false);
}

// One workgroup (256 threads = 8 wave32) processes ROWS_PER_WG batch rows.
// LDS: z ping-pong buffers + stage-A twiddle table. 48 KB total.
__global__ __launch_bounds__(TPB) void phasor_fft_kernel(
    const float* __restrict__ x, const float* __restrict__ w,
    float* __restrict__ out, int batch) {
  __shared__ float bufR[SEQ], bufI[SEQ];
  __shared__ float tmpR[SEQ], tmpI[SEQ];
  __shared__ float twR[SEQ], twI[SEQ];

  const int t    = threadIdx.x;
  const int lane = t & 31;
  const int wv   = t >> 5;      // wave id 0..7
  const int half = lane >> 4;   // 0/1 (lane group)
  const int l16  = lane & 15;

  // ---- DFT-16 matrix as WMMA A-fragments, resident in VGPRs all kernel.
  // A(16x4) layout: lane L holds row M=L%16; VGPR j holds K = 4*kk + 2*(L/16)+j.
  // AiN = -Ai so the Re-accumulator chain needs no per-iteration negation.
  v2f Ar[4], Ai[4], AiN[4];
  {
    const float w16 = -2.0f * PI_F / 16.0f;
#pragma unroll
    for (int kk = 0; kk < 4; ++kk) {
#pragma unroll
      for (int j = 0; j < 2; ++j) {
        const int K = 4 * kk + 2 * half + j;
        float s, c;
        __sincosf(w16 * (float)((l16 * K) & 15), &s, &c);
        Ar[kk][j]  = c;
        Ai[kk][j]  = s;
        AiN[kk][j] = -s;
      }
    }
  }

  // ---- Stage-A twiddle table W_2048^{k1*n2} at [128*k1 + n2], with the
  // ortho 1/sqrt(2048) scale folded in (applied exactly once per FFT).
  {
    const float wbig = -2.0f * PI_F / 2048.0f;
    const float invs = 0.02209708691207961f;  // 1/sqrt(2048)
#pragma unroll
    for (int i = 0; i < SEQ / TPB; ++i) {
      const int n = t + TPB * i;
      const int k1 = n >> 7, n2 = n & 127;
      float s, c;
      __sincosf(wbig * (float)(k1 * n2), &s, &c);
      twR[n] = c * invs;
      twI[n] = s * invs;
    }
  }
  __syncthreads();

  for (int rr = 0; rr < ROWS_PER_WG; ++rr) {
    const int row = blockIdx.x * ROWS_PER_WG + rr;
    if (row >= batch) break;

    // Prefetch next row's input (global_prefetch_b8): 256 lanes x 32B = 8KB row.
    if (rr + 1 < ROWS_PER_WG && row + 1 < batch) {
      __builtin_prefetch(x + (size_t)(row + 1) * SEQ + t * 8, 0, 3);
    }

    // ---- z = exp(i*(x + pre-shift of block 0)).  w layout: w[(2b+p)*SEQ+n].
    {
      const float* xr = x + (size_t)row * SEQ;
#pragma unroll
      for (int i = 0; i < SEQ / TPB; ++i) {
        const int n = t + TPB * i;
        float s, c;
        __sincosf(xr[n] + w[n], &s, &c);
        bufR[n] = c;
        bufI[n] = s;
      }
    }
    __syncthreads();

    for (int b = 0; b < 4; ++b) {
      // ===== Stage A: Y[k1][n2] = W2048^{k1 n2}/sqrt(N) *
      //                sum_n1 D16[k1][n1] z[128 n1 + n2]      (buf -> tmp)
      // 8 waves each own a 16-column tile of n2.
      {
        const int n2 = 16 * wv + l16;
        v2f Br[4], Bi[4];
#pragma unroll
        for (int kk = 0; kk < 4; ++kk) {
#pragma unroll
          for (int j = 0; j < 2; ++j) {
            const int K = 4 * kk + 2 * half + j;  // n1
            Br[kk][j] = bufR[128 * K + n2];
            Bi[kk][j] = bufI[128 * K + n2];
          }
        }
        v8f cr = {}, ci = {};
#pragma unroll
        for (int kk = 0; kk < 4; ++kk) cr = wmma4(Ar[kk], Br[kk], cr);
#pragma unroll
        for (int kk = 0; kk < 4; ++kk) cr = wmma4(AiN[kk], Bi[kk], cr);
#pragma unroll
        for (int kk = 0; kk < 4; ++kk) ci = wmma4(Ar[kk], Bi[kk], ci);
#pragma unroll
        for (int kk = 0; kk < 4; ++kk) ci = wmma4(Ai[kk], Br[kk], ci);
        // C/D layout: VGPR r -> row M = r + 8*(lane/16), col N = lane%16.
#pragma unroll
        for (int r = 0; r < 8; ++r) {
          const int k1 = r + 8 * half;
          const int idx = 128 * k1 + n2;
          const float yr = cr[r], yi = ci[r];
          const float c = twR[idx], s = twI[idx];
          tmpR[idx] = yr * c - yi * s;
          tmpI[idx] = yr * s + yi * c;
        }
      }
      __syncthreads();

      // ===== Stage B1: per k1-row DFT-16 over a (n2 = 8a + b), two k1 rows
      // packed per 16x16 tile.  tmp -> buf, twiddle W_128^{b*d} on store.
      {
        const int k1 = 2 * wv + (l16 >> 3);
        const int bb = l16 & 7;
        v2f Br[4], Bi[4];
#pragma unroll
        for (int kk = 0; kk < 4; ++kk) {
#pragma unroll
          for (int j = 0; j < 2; ++j) {
            const int a = 4 * kk + 2 * half + j;
            Br[kk][j] = tmpR[128 * k1 + 8 * a + bb];
            Bi[kk][j] = tmpI[128 * k1 + 8 * a + bb];
          }
        }
        v8f cr = {}, ci = {};
#pragma unroll
        for (int kk = 0; kk < 4; ++kk) cr = wmma4(Ar[kk], Br[kk], cr);
#pragma unroll
        for (int kk = 0; kk < 4; ++kk) cr = wmma4(AiN[kk], Bi[kk], cr);
#pragma unroll
        for (int kk = 0; kk < 4; ++kk) ci = wmma4(Ar[kk], Bi[kk], ci);
#pragma unroll
        for (int kk = 0; kk < 4; ++kk) ci = wmma4(Ai[kk], Br[kk], ci);
        const float w128 = -2.0f * PI_F / 128.0f;
#pragma unroll
        for (int r = 0; r < 8; ++r) {
          const int d = r + 8 * half;
          float s, c;
          __sincosf(w128 * (float)(bb * d), &s, &c);
          const float yr = cr[r], yi = ci[r];
          bufR[128 * k1 + 8 * d + bb] = yr * c - yi * s;
          bufI[128 * k1 + 8 * d + bb] = yr * s + yi * c;
        }
      }
      __syncthreads();

      // ===== Stage B3: 256 independent 8-point FFTs (VALU butterflies).
      // Thread (k1,d) reads g[b]=buf[128 k1 + 8 d + b], writes
      // Z[c] -> tmp[k1 + 16 d + 256 c]   (full FFT index k = k1 + 16*k2).
      {
        const int k1 = t >> 4;
        const int d  = t & 15;
        const int bi = 128 * k1 + 8 * d;
        float gr[8], gi[8];
#pragma unroll
        for (int q = 0; q < 8; ++q) {
          gr[q] = bufR[bi + q];
          gi[q] = bufI[bi + q];
        }
        const float C = 0.70710678118654752f;
        // radix-2 DIF
        const float a0r = gr[0] + gr[4], a0i = gi[0] + gi[4];
        const float a1r = gr[1] + gr[5], a1i = gi[1] + gi[5];
        const float a2r = gr[2] + gr[6], a2i = gi[2] + gi[6];
        const float a3r = gr[3] + gr[7], a3i = gi[3] + gi[7];
        const float b0r = gr[0] - gr[4], b0i = gi[0] - gi[4];
        const float t1r = gr[1] - gr[5], t1i = gi[1] - gi[5];
        const float t2r = gr[2] - gr[6], t2i = gi[2] - gi[6];
        const float t3r = gr[3] - gr[7], t3i = gi[3] - gi[7];
        const float b1r = C * (t1r + t1i), b1i = C * (t1i - t1r);   // *W8^1
        const float b2r = t2i,             b2i = -t2r;              // *W8^2
        const float b3r = C * (t3i - t3r), b3i = -C * (t3r + t3i);  // *W8^3
        // even half (FFT4 of a)
        const float c0r = a0r + a2r, c0i = a0i + a2i;
        const float c1r = a1r + a3r, c1i = a1i + a3i;
        const float d0r = a0r - a2r, d0i = a0i - a2i;
        const float d1r = a1i - a3i, d1i = -(a1r - a3r);            // *(-i)
        // odd half (FFT4 of b)
        const float e0r = b0r + b2r, e0i = b0i + b2i;
        const float e1r = b1r + b3r, e1i = b1i + b3i;
        const float f0r = b0r - b2r, f0i = b0i - b2i;
        const float f1r = b1i - b3i, f1i = -(b1r - b3r);            // *(-i)
        const int base = k1 + 16 * d;
        tmpR[base + 0 * 256] = c0r + c1r; tmpI[base + 0 * 256] = c0i + c1i;
        tmpR[base + 1 * 256] = e0r + e1r; tmpI[base + 1 * 256] = e0i + e1i;
        tmpR[base + 2 * 256] = d0r + d1r; tmpI[base + 2 * 256] = d0i + d1i;
        tmpR[base + 3 * 256] = f0r + f1r; tmpI[base + 3 * 256] = f0i + f1i;
        tmpR[base + 4 * 256] = c0r - c1r; tmpI[base + 4 * 256] = c0i - c1i;
        tmpR[base + 5 * 256] = e0r - e1r; tmpI[base + 5 * 256] = e0i - e1i;
        tmpR[base + 6 * 256] = d0r - d1r; tmpI[base + 6 * 256] = d0i - d1i;
        tmpR[base + 7 * 256] = f0r - f1r; tmpI[base + 7 * 256] = f0i - f1i;
      }
      __syncthreads();

      // ===== Inter-block rotation: post[b] + pre[b+1] (scale already folded
      // into the stage-A twiddle table).
      if (b < 3) {
#pragma unroll
        for (int i = 0; i < SEQ / TPB; ++i) {
          const int n = t + TPB * i;
          const float ph = w[(2 * b + 1) * SEQ + n] + w[(2 * b + 2) * SEQ + n];
          float s, c;
          __sincosf(ph, &s, &c);
          const float zr = tmpR[n], zi = tmpI[n];
          bufR[n] = zr * c - zi * s;
          bufI[n] = zr * s + zi * c;
        }
        __syncthreads();
      }
    }

    // ===== Readout: phase of z[0] (+ last post shift), wrapped by asin(sin).
    // (tmp is not overwritten until the next row's Stage A, which is behind a
    // barrier after the next load pass.)
    if (t == 0) {
      const float zr = tmpR[0], zi = tmpI[0];
      const float ph = atan2f(zi, zr) + w[7 * SEQ];
      out[row] = asinf(sinf(ph));
    }
  }
}

extern "C" void kernel_launch(void* const* d_in, const int* in_sizes, int n_in,
                              void* d_out, int out_size, void* d_ws,
                              size_t ws_size, hipStream_t stream) {
  (void)n_in; (void)out_size; (void)d_ws; (void)ws_size;
  const float* x = (const float*)d_in[0];
  const float* w = (const float*)d_in[1];
  float* out = (float*)d_out;
  const int batch = in_sizes[0] / SEQ;
  const int grid = (batch + ROWS_PER_WG - 1) / ROWS_PER_WG;
  phasor_fft_kernel<<<dim3(grid), dim3(TPB), 0, stream>>>(x, w, out, batch);
}